// NodeEmbedder_87797721465321
// MI455X (gfx1250) — compile-verified
//
#include <hip/hip_runtime.h>
#include <math.h>

typedef __attribute__((ext_vector_type(2))) float v2f;
typedef __attribute__((ext_vector_type(8))) float v8f;

#define EPSF  1e-15f
#define SQC   0.1f          /* sqrt(c), c = 0.01 */
#define CC    0.01f
#define MAXN  9.96f         /* (1 - 4e-3)/sqrt(c) */
#define CLIP1 (1.0f - 1e-7f)

// ---------------- helpers ----------------
__device__ __forceinline__ float hsum16(float v) {   // reduce within 16-lane half
    v += __shfl_xor(v, 1);
    v += __shfl_xor(v, 2);
    v += __shfl_xor(v, 4);
    v += __shfl_xor(v, 8);
    return v;
}
__device__ __forceinline__ float hsum32(float v) {   // reduce across wave32
    v = hsum16(v);
    v += __shfl_xor(v, 16);
    return v;
}
// order-preserving float->uint key for atomicMax over signed floats
__device__ __forceinline__ unsigned fenc(float f) {
    unsigned u = __float_as_uint(f);
    return (u & 0x80000000u) ? ~u : (u | 0x80000000u);
}
__device__ __forceinline__ float fdec(unsigned k) {
    unsigned u = (k & 0x80000000u) ? (k & 0x7fffffffu) : ~k;
    return __uint_as_float(u);
}
#define NEGINF_KEY 0x007fffffu   /* fenc(-inf) */

__device__ __forceinline__ float dot4(float4 a, float4 b) {
    return a.x * b.x + a.y * b.y + a.z * b.z + a.w * b.w;
}

// ---------------- 0: workspace init (ws is poisoned once, never re-poisoned) --
__global__ void init_kernel(unsigned* __restrict__ mkey, float* __restrict__ ssum,
                            float* __restrict__ sup, int nm, int nsup) {
    int i = blockIdx.x * 256 + threadIdx.x;
    if (i < nsup) sup[i] = 0.0f;
    if (i < nm) { mkey[i] = NEGINF_KEY; ssum[i] = 0.0f; }
}

// ---------------- 1: xt = logmap0(projx(expmap0(emb))), one wave per node ----
__global__ void xt_kernel(const float* __restrict__ emb, float* __restrict__ xt, int N) {
    int wave = threadIdx.x >> 5, lane = threadIdx.x & 31;
    int node = blockIdx.x * 8 + wave;
    if (node >= N) return;                       // wave-uniform branch
    float2 v = ((const float2*)(emb + (size_t)node * 64))[lane];
    float n0 = fmaxf(sqrtf(hsum32(v.x * v.x + v.y * v.y)), EPSF);
    float se = tanhf(SQC * n0) / (SQC * n0);     // expmap0 scale
    float n1 = fmaxf(se * n0, EPSF);
    float sp = (n1 > MAXN) ? (MAXN / n1) : 1.0f; // projx
    float n2 = fmaxf(fminf(n1, MAXN), EPSF);
    float t  = fminf(SQC * n2, CLIP1);
    float sl = atanhf(t) / (SQC * n2);           // logmap0 scale
    float s  = se * sp * sl;
    float2 o; o.x = v.x * s; o.y = v.y * s;
    ((float2*)(xt + (size_t)node * 64))[lane] = o;
}

// ---------------- 2: alpha + leaky_relu + segment max, thread = (edge,head) --
__global__ void alpha_kernel(const float* __restrict__ xt, const int* __restrict__ ei,
                             const float* __restrict__ atti, const float* __restrict__ attj,
                             float* __restrict__ aBuf, unsigned* __restrict__ mkey,
                             int N, int E, int E2, int S) {
    int gi  = blockIdx.y;
    int idx = blockIdx.x * 256 + threadIdx.x;
    if (idx >= E2 * 4) return;
    int e = idx >> 2, h = idx & 3;
    size_t gb = (size_t)(gi * S + (S - 1)) * 2 * E;
    int src = (e < E) ? ei[gb + e]     : (e - E);
    int dst = (e < E) ? ei[gb + E + e] : (e - E);
    const float4* xi = (const float4*)(xt + (size_t)src * 64 + h * 16);
    const float4* xj = (const float4*)(xt + (size_t)dst * 64 + h * 16);
    const float4* ai = (const float4*)(atti + h * 16);
    const float4* aj = (const float4*)(attj + h * 16);
    float a = 0.0f;
#pragma unroll
    for (int q = 0; q < 4; ++q) a += dot4(xi[q], ai[q]) + dot4(xj[q], aj[q]);
    a = (a >= 0.0f) ? a : 0.2f * a;              // leaky_relu(0.2)
    aBuf[(size_t)gi * E2 * 4 + idx] = a;
    atomicMax(&mkey[((size_t)gi * N + src) * 4 + h], fenc(a));
}

// ---------------- 3: a = exp(alpha - m[src]); segment sum ------------------
__global__ void expsum_kernel(const int* __restrict__ ei, const unsigned* __restrict__ mkey,
                              float* __restrict__ aBuf, float* __restrict__ ssum,
                              int N, int E, int E2, int S) {
    int gi  = blockIdx.y;
    int idx = blockIdx.x * 256 + threadIdx.x;
    if (idx >= E2 * 4) return;
    int e = idx >> 2, h = idx & 3;
    size_t gb = (size_t)(gi * S + (S - 1)) * 2 * E;
    int src = (e < E) ? ei[gb + e] : (e - E);
    float m = fdec(mkey[((size_t)gi * N + src) * 4 + h]);
    float a = expf(aBuf[(size_t)gi * E2 * 4 + idx] - m);
    aBuf[(size_t)gi * E2 * 4 + idx] = a;
    atomicAdd(&ssum[((size_t)gi * N + src) * 4 + h], a);
}

// ---------------- 4: sup[src] += mean_h( a_h/s_h * x_j[h] ), thread=(edge,c) -
__global__ void sup_kernel(const float* __restrict__ xt, const int* __restrict__ ei,
                           const float* __restrict__ aBuf, const float* __restrict__ ssum,
                           float* __restrict__ sup, int N, int E, int E2, int S) {
    int gi  = blockIdx.y;
    long long idx = (long long)blockIdx.x * 256 + threadIdx.x;
    if (idx >= (long long)E2 * 16) return;
    int e = (int)(idx >> 4), c = (int)(idx & 15);
    size_t gb = (size_t)(gi * S + (S - 1)) * 2 * E;
    int src = (e < E) ? ei[gb + e]     : (e - E);
    int dst = (e < E) ? ei[gb + E + e] : (e - E);
    float4 av = *(const float4*)(aBuf + ((size_t)gi * E2 + e) * 4);
    float4 sv = *(const float4*)(ssum + ((size_t)gi * N + src) * 4);
    const float* xj = xt + (size_t)dst * 64;
    float val = av.x / (sv.x + 1e-16f) * xj[c]
              + av.y / (sv.y + 1e-16f) * xj[16 + c]
              + av.z / (sv.z + 1e-16f) * xj[32 + c]
              + av.w / (sv.w + 1e-16f) * xj[48 + c];
    atomicAdd(&sup[((size_t)gi * N + src) * 16 + c], 0.25f * val);
}

// ---------------- 5: node_emb = projx(expmap0(sup)), thread per row ---------
__global__ void nodeemb_kernel(const float* __restrict__ sup, float* __restrict__ ne, int rows) {
    int r = blockIdx.x * 256 + threadIdx.x;
    if (r >= rows) return;
    const float4* p = (const float4*)(sup + (size_t)r * 16);
    float4 v[4]; float ss = 0.0f;
#pragma unroll
    for (int q = 0; q < 4; ++q) { v[q] = p[q]; ss += dot4(v[q], v[q]); }
    float n  = fmaxf(sqrtf(ss), EPSF);
    float se = tanhf(SQC * n) / (SQC * n);
    float n1 = fmaxf(se * n, EPSF);
    float s  = se * ((n1 > MAXN) ? (MAXN / n1) : 1.0f);
    float4* o = (float4*)(ne + (size_t)r * 16);
#pragma unroll
    for (int q = 0; q < 4; ++q) {
        float4 w; w.x = v[q].x * s; w.y = v[q].y * s; w.z = v[q].z * s; w.w = v[q].w * s;
        o[q] = w;
    }
}

// ---------------- 6: hb = projx(expmap0(b)) for b1, b2 (64 threads) ---------
// layout: hb[0..63]=hb1, hb[64..127]=hb2, hb[128]=||hb1||^2, hb[129]=||hb2||^2
__global__ void hb_kernel(const float* __restrict__ b1, const float* __restrict__ b2,
                          float* __restrict__ hb) {
    int wave = threadIdx.x >> 5, lane = threadIdx.x & 31;
    const float* b = (wave == 0) ? b1 : b2;
    float2 v = ((const float2*)b)[lane];
    float n0 = fmaxf(sqrtf(hsum32(v.x * v.x + v.y * v.y)), EPSF);
    float se = tanhf(SQC * n0) / (SQC * n0);
    float n1 = fmaxf(se * n0, EPSF);
    float sp = (n1 > MAXN) ? (MAXN / n1) : 1.0f;
    float s  = se * sp;
    float2 o; o.x = v.x * s; o.y = v.y * s;
    ((float2*)(hb + wave * 64))[lane] = o;
    float nn = fminf(n1, MAXN);
    if (lane == 0) hb[128 + wave] = nn * nn;
}

// ---------------- 7: hyperbolic MLP via f32 WMMA ----------------------------
// Each wave owns 16 rows: GEMM1 [16x16]@[16x64] (16 wmma), mobius chain,
// SiLU/log-exp chain, GEMM2 [16x64]@[64x64] (64 wmma), mobius chain, store.
__global__ __launch_bounds__(256) void mlp_kernel(
        const float* __restrict__ ne, const float* __restrict__ w1,
        const float* __restrict__ w2, const float* __restrict__ hb,
        float* __restrict__ out, int M) {
    __shared__ float tile[8][16][68];
    __shared__ float rsc[8][16];
    const int wave = threadIdx.x >> 5, lane = threadIdx.x & 31;
    const int half = lane >> 4, lr = lane & 15;
    const int rowBase = blockIdx.x * 128 + wave * 16;

    // ---- A fragments for GEMM1 (K=16): lane l<16 row=lr cols 4kk+{0,1};
    //      lane>=16 same rows, cols 4kk+{2,3}  (ISA 16x4 f32 A layout)
    v2f a1[4]; float xp = 0.0f;
#pragma unroll
    for (int kk = 0; kk < 4; ++kk) {
        int r = rowBase + lr;
        int rc = (r < M) ? r : 0;
        float ok = (r < M) ? 1.0f : 0.0f;
        const float* p = ne + (size_t)rc * 16 + kk * 4 + half * 2;
        float e0 = p[0] * ok, e1 = p[1] * ok;
        a1[kk].x = e0; a1[kk].y = e1;
        xp += e0 * e0 + e1 * e1;
    }
    {
        float xn2 = xp + __shfl_xor(xp, 16);     // full ||x_row||^2, row = lr
        if (half == 0) rsc[wave][lr] = xn2;
    }
    __syncthreads();

    // ---- GEMM1: mx = x @ w1^T  (B[k][n] = w1[n][k])
    v8f acc[4];
#pragma unroll
    for (int nt = 0; nt < 4; ++nt) {
        v8f c = {0.f, 0.f, 0.f, 0.f, 0.f, 0.f, 0.f, 0.f};
#pragma unroll
        for (int kk = 0; kk < 4; ++kk) {
            const float* wp = w1 + (size_t)(nt * 16 + lr) * 16 + kk * 4 + half * 2;
            v2f b; b.x = wp[0]; b.y = wp[1];
            c = __builtin_amdgcn_wmma_f32_16x16x4_f32(false, a1[kk], false, b,
                                                      (short)0, c, false, false);
        }
        acc[nt] = c;
    }

    // ---- layer-1 post: mobius_matvec scale, projx, mobius_add(hb1), projx
    float z[4][8], rn[8];
    float hreg[4];
#pragma unroll
    for (int nt = 0; nt < 4; ++nt) hreg[nt] = hb[nt * 16 + lr];
    float y2 = hb[128];
#pragma unroll
    for (int v = 0; v < 8; ++v) {
        int vrow = v + 8 * half;                   // C/D layout: row = v + 8*half
        float xn = fmaxf(sqrtf(rsc[wave][vrow]), EPSF);
        float p = 0.0f;
#pragma unroll
        for (int nt = 0; nt < 4; ++nt) { float e = acc[nt][v]; p += e * e; }
        float mx2 = hsum16(p);
        float mxn = fmaxf(sqrtf(mx2), EPSF);
        float t = fminf(SQC * xn, CLIP1);
        float g = tanhf(mxn / xn * atanhf(t));
        float sc = g / (mxn * SQC);
        float nrm = g / SQC;
        if (mx2 == 0.0f) { sc = 0.0f; nrm = 0.0f; }
        if (nrm > MAXN) { sc *= MAXN / nrm; nrm = MAXN; }
        float xy = 0.0f;
#pragma unroll
        for (int nt = 0; nt < 4; ++nt) xy += (sc * acc[nt][v]) * hreg[nt];
        xy = hsum16(xy);
        float x2 = nrm * nrm;
        float den = fmaxf(1.0f + 2.0f * CC * xy + CC * CC * x2 * y2, EPSF);
        float cx = (1.0f + 2.0f * CC * xy + CC * y2) / den;
        float cy = (1.0f - CC * x2) / den;
        float o2 = 0.0f;
#pragma unroll
        for (int nt = 0; nt < 4; ++nt) {
            float o = cx * sc * acc[nt][v] + cy * hreg[nt];
            z[nt][v] = o; o2 += o * o;
        }
        o2 = hsum16(o2);
        float on = fmaxf(sqrtf(o2), EPSF);
        float ps = (on > MAXN) ? (MAXN / on) : 1.0f;
#pragma unroll
        for (int nt = 0; nt < 4; ++nt) z[nt][v] *= ps;
        rn[v] = fminf(on, MAXN);
    }

    // ---- HypAct(SiLU) + HypDropout(eval) chain
#pragma unroll
    for (int v = 0; v < 8; ++v) {
        float n1 = fmaxf(rn[v], EPSF);
        float ls = atanhf(fminf(SQC * n1, CLIP1)) / (SQC * n1);   // logmap0
        float p = 0.0f;
#pragma unroll
        for (int nt = 0; nt < 4; ++nt) {
            float u = ls * z[nt][v];
            float s = u / (1.0f + expf(-u));                       // silu
            z[nt][v] = s; p += s * s;
        }
        float n2 = fmaxf(sqrtf(hsum16(p)), EPSF);
        float es = tanhf(SQC * n2) / (SQC * n2);                   // expmap0
        float nrm = tanhf(SQC * n2) / SQC;
        float sc = es;
        if (nrm > MAXN) { sc *= MAXN / nrm; nrm = MAXN; }          // projx
        float n3 = fmaxf(nrm, EPSF);                               // log/exp round trip
        float t3 = fminf(SQC * n3, CLIP1);
        float l3 = atanhf(t3) / (SQC * n3);
        float un = fmaxf(atanhf(t3) / SQC, EPSF);
        float e3 = tanhf(SQC * un) / (SQC * un);
        float nrm2 = tanhf(SQC * un) / SQC;
        float sc2 = l3 * e3;
        if (nrm2 > MAXN) { sc2 *= MAXN / nrm2; nrm2 = MAXN; }      // projx
        float st = sc * sc2;
#pragma unroll
        for (int nt = 0; nt < 4; ++nt) z[nt][v] *= st;
        rn[v] = nrm2;
    }

    // ---- stage z3 + its norms to LDS (register-layout transpose for GEMM2)
    if (lr == 0) {
#pragma unroll
        for (int v = 0; v < 8; ++v) rsc[wave][v + 8 * half] = rn[v];
    }
#pragma unroll
    for (int nt = 0; nt < 4; ++nt)
#pragma unroll
        for (int v = 0; v < 8; ++v)
            tile[wave][v + 8 * half][nt * 16 + lr] = z[nt][v];
    __syncthreads();

    // ---- GEMM2: mx2 = z3 @ w2^T  (K = 64)
    v2f a2[16];
#pragma unroll
    for (int kk = 0; kk < 16; ++kk) {
        a2[kk].x = tile[wave][lr][kk * 4 + half * 2];
        a2[kk].y = tile[wave][lr][kk * 4 + half * 2 + 1];
    }
    v8f acc2[4];
#pragma unroll
    for (int nt = 0; nt < 4; ++nt) {
        v8f c = {0.f, 0.f, 0.f, 0.f, 0.f, 0.f, 0.f, 0.f};
#pragma unroll
        for (int kk = 0; kk < 16; ++kk) {
            const float* wp = w2 + (size_t)(nt * 16 + lr) * 64 + kk * 4 + half * 2;
            v2f b; b.x = wp[0]; b.y = wp[1];
            c = __builtin_amdgcn_wmma_f32_16x16x4_f32(false, a2[kk], false, b,
                                                      (short)0, c, false, false);
        }
        acc2[nt] = c;
    }

    // ---- layer-2 post + store
#pragma unroll
    for (int nt = 0; nt < 4; ++nt) hreg[nt] = hb[64 + nt * 16 + lr];
    float y2b = hb[129];
#pragma unroll
    for (int v = 0; v < 8; ++v) {
        int vrow = v + 8 * half;
        float xn = fmaxf(rsc[wave][vrow], EPSF);
        float p = 0.0f;
#pragma unroll
        for (int nt = 0; nt < 4; ++nt) { float e = acc2[nt][v]; p += e * e; }
        float mx2 = hsum16(p);
        float mxn = fmaxf(sqrtf(mx2), EPSF);
        float t = fminf(SQC * xn, CLIP1);
        float g = tanhf(mxn / xn * atanhf(t));
        float sc = g / (mxn * SQC);
        float nrm = g / SQC;
        if (mx2 == 0.0f) { sc = 0.0f; nrm = 0.0f; }
        if (nrm > MAXN) { sc *= MAXN / nrm; nrm = MAXN; }
        float xy = 0.0f;
#pragma unroll
        for (int nt = 0; nt < 4; ++nt) xy += (sc * acc2[nt][v]) * hreg[nt];
        xy = hsum16(xy);
        float x2 = nrm * nrm;
        float den = fmaxf(1.0f + 2.0f * CC * xy + CC * CC * x2 * y2b, EPSF);
        float cx = (1.0f + 2.0f * CC * xy + CC * y2b) / den;
        float cy = (1.0f - CC * x2) / den;
        float o2 = 0.0f;
        float ov[4];
#pragma unroll
        for (int nt = 0; nt < 4; ++nt) {
            float o = cx * sc * acc2[nt][v] + cy * hreg[nt];
            ov[nt] = o; o2 += o * o;
        }
        o2 = hsum16(o2);
        float on = fmaxf(sqrtf(o2), EPSF);
        float ps = (on > MAXN) ? (MAXN / on) : 1.0f;
        int r = rowBase + vrow;
        if (r < M) {
#pragma unroll
            for (int nt = 0; nt < 4; ++nt)
                out[(size_t)r * 64 + nt * 16 + lr] = ov[nt] * ps;
        }
    }
}

// ---------------- host launcher ---------------------------------------------
extern "C" void kernel_launch(void* const* d_in, const int* in_sizes, int n_in,
                              void* d_out, int out_size, void* d_ws, size_t ws_size,
                              hipStream_t stream) {
    if (n_in < 9) return;
    const int*   ei   = (const int*)d_in[1];
    const float* emb  = (const float*)d_in[2];
    const float* atti = (const float*)d_in[3];
    const float* attj = (const float*)d_in[4];
    const float* w1   = (const float*)d_in[5];
    const float* b1   = (const float*)d_in[6];
    const float* w2   = (const float*)d_in[7];
    const float* b2   = (const float*)d_in[8];
    float* out = (float*)d_out;

    const int H = 64;
    const int N = in_sizes[2] / H;            // 10000
    const int G = in_sizes[0] / N;            // B*S = 8
    const int Bb = out_size / (N * H);        // 2
    const int S = G / Bb;                     // 4
    const int E = in_sizes[1] / (G * 2);      // 160000
    const int E2 = E + N;                     // with self loops
    const int M = Bb * N;                     // 20000 output rows

    // workspace layout (floats)
    float* ws = (float*)d_ws;
    float*    xt    = ws;                                   // N*64
    float*    aBuf  = xt + (size_t)N * 64;                  // Bb*E2*4
    unsigned* mkey  = (unsigned*)(aBuf + (size_t)Bb * E2 * 4); // Bb*N*4
    float*    ssum  = (float*)(mkey + (size_t)Bb * N * 4);  // Bb*N*4
    float*    sup   = ssum + (size_t)Bb * N * 4;            // M*16
    float*    neBuf = sup + (size_t)M * 16;                 // M*16
    float*    hbuf  = neBuf + (size_t)M * 16;               // 130
    size_t need = (size_t)(hbuf - ws) + 130;
    if (ws_size < need * sizeof(float)) return;

    int nm = Bb * N * 4, nsup = M * 16;
    int nInit = nsup > nm ? nsup : nm;
    init_kernel<<<(nInit + 255) / 256, 256, 0, stream>>>(mkey, ssum, sup, nm, nsup);

    xt_kernel<<<(N + 7) / 8, 256, 0, stream>>>(emb, xt, N);

    dim3 ga((E2 * 4 + 255) / 256, Bb);
    alpha_kernel<<<ga, 256, 0, stream>>>(xt, ei, atti, attj, aBuf, mkey, N, E, E2, S);
    expsum_kernel<<<ga, 256, 0, stream>>>(ei, mkey, aBuf, ssum, N, E, E2, S);

    dim3 gs((unsigned)(((long long)E2 * 16 + 255) / 256), Bb);
    sup_kernel<<<gs, 256, 0, stream>>>(xt, ei, aBuf, ssum, sup, N, E, E2, S);

    nodeemb_kernel<<<(M + 255) / 256, 256, 0, stream>>>(sup, neBuf, M);
    hb_kernel<<<1, 64, 0, stream>>>(b1, b2, hbuf);

    mlp_kernel<<<(M + 127) / 128, 256, 0, stream>>>(neBuf, w1, w2, hbuf, out, M);
}